// EdgeToEdgeElementWis_5420248727808
// MI455X (gfx1250) — compile-verified
//
#include <hip/hip_runtime.h>

typedef __attribute__((ext_vector_type(2))) float v2f;
typedef __attribute__((ext_vector_type(4))) float v4f;
typedef __attribute__((ext_vector_type(8))) float v8f;

#define DIM 90
#define DD  8100      // 90*90
#define NB  128       // N
#define PB  64        // P

// ---------------------------------------------------------------------------
// Phase 1: compute colsum[n,p,j] and rowsum[n,p,i] with V_WMMA_F32_16X16X4_F32.
// Each wave owns one (mode, slice, n-tile16, p-tile16) and runs K=90 in steps
// of 4 (zero-padded tail). 5760 waves total = 720 blocks * 8 waves, so every
// wave is fully populated (EXEC all-1s, as WMMA requires).
// ---------------------------------------------------------------------------
__global__ __launch_bounds__(256) void e2e_reduce_wmma(
    const float* __restrict__ x, const float* __restrict__ w,
    float* __restrict__ ws) {
  const int lane = threadIdx.x & 31;
  int wv = blockIdx.x * 8 + (threadIdx.x >> 5);
  const int pt = wv & 3;  wv >>= 2;         // p-tile 0..3
  const int nt = wv & 7;  wv >>= 3;         // n-tile 0..7
  const int slice = wv % DIM;               // fixed j (mode 0) or fixed i (mode 1)
  const int mode  = wv / DIM;               // 0: colsum, 1: rowsum

  const int s   = mode ? 1 : DIM;           // stride along K inside one image
  const int off = mode ? slice * DIM : slice;

  const int hi = lane >> 4;                 // lane half selects K pair {0,1}/{2,3}
  const int l  = lane & 15;

  // A fragment rows: n = nt*16 + l ; B fragment cols: p = pt*16 + l
  const float* xa = x + (size_t)(nt * 16 + l) * DD + off;
  const float* wb = w + (size_t)(pt * 16 + l) * DD + off;

  v8f c = {};
  for (int k0 = 0; k0 < DIM; k0 += 4) {
    const int ka = k0 + hi * 2;
    const int kb = ka + 1;
    v2f a, b;
    a.x = (ka < DIM) ? xa[ka * s] : 0.0f;
    a.y = (kb < DIM) ? xa[kb * s] : 0.0f;
    b.x = (ka < DIM) ? wb[ka * s] : 0.0f;
    b.y = (kb < DIM) ? wb[kb * s] : 0.0f;
    // 8 args: (neg_a, A, neg_b, B, c_mod, C, reuse_a, reuse_b)
    c = __builtin_amdgcn_wmma_f32_16x16x4_f32(false, a, false, b, (short)0, c,
                                              false, false);
  }

  // C/D layout: VGPR r -> row M = r (+8 for upper lane half); N = lane&15.
  float* ob = ws + (size_t)mode * (NB * PB * DIM);
  const int p = pt * 16 + l;
#pragma unroll
  for (int r = 0; r < 8; ++r) {
    const int n = nt * 16 + hi * 8 + r;
    ob[((size_t)n * PB + p) * DIM + slice] = c[r];
  }
}

// ---------------------------------------------------------------------------
// Phase 2: out[n,p,i,j] = colsum[n,p,j] + rowsum[n,p,i].
// Store-bandwidth bound (265 MB). One block per (n,p): stage the two 90-float
// rows in LDS, then emit the contiguous 8100-float chunk as 2025 x 128-bit
// non-temporal stores (output exceeds L2 and is never re-read).
// ---------------------------------------------------------------------------
__global__ __launch_bounds__(256) void e2e_broadcast_add(
    const float* __restrict__ ws, float* __restrict__ out) {
  __shared__ float cs[DIM];
  __shared__ float rs[DIM];
  const int np = blockIdx.x;                // 0 .. 128*64-1
  const int t  = threadIdx.x;

  const float* colsum = ws;
  const float* rowsum = ws + (size_t)NB * PB * DIM;
  if (t < DIM)                       cs[t]       = colsum[(size_t)np * DIM + t];
  else if (t >= 128 && t < 128+DIM)  rs[t - 128] = rowsum[(size_t)np * DIM + (t - 128)];
  __syncthreads();

  v4f* o = (v4f*)(out + (size_t)np * DD);
  for (int v = t; v < DD / 4; v += 256) {     // 2025 float4 per (n,p) chunk
    const int e = v * 4;
    int i = e / DIM;
    int j = e - i * DIM;
    v4f r;
    r.x = rs[i] + cs[j]; if (++j == DIM) { j = 0; ++i; }
    r.y = rs[i] + cs[j]; if (++j == DIM) { j = 0; ++i; }
    r.z = rs[i] + cs[j]; if (++j == DIM) { j = 0; ++i; }
    r.w = rs[i] + cs[j];
    __builtin_nontemporal_store(r, o + v);
  }
}

// ---------------------------------------------------------------------------
extern "C" void kernel_launch(void* const* d_in, const int* in_sizes, int n_in,
                              void* d_out, int out_size, void* d_ws, size_t ws_size,
                              hipStream_t stream) {
  const float* x = (const float*)d_in[0];   // (128,1,90,90)
  const float* w = (const float*)d_in[1];   // (64,1,90,90)
  float* out = (float*)d_out;               // (128,64,90,90)
  float* ws  = (float*)d_ws;                // 2 * 128*64*90 floats = 5.9 MB

  // 2 modes * 90 slices * 8 n-tiles * 4 p-tiles = 5760 waves -> 720 blocks
  e2e_reduce_wmma<<<720, 256, 0, stream>>>(x, w, ws);
  // one block per (n,p) pair
  e2e_broadcast_add<<<NB * PB, 256, 0, stream>>>(ws, out);
}